// Model_38465727103247
// MI455X (gfx1250) — compile-verified
//
#include <hip/hip_runtime.h>
#include <math.h>

// ---------------- problem constants (from reference) ----------------
#define KG_NUM_ENTITY   14952
#define KG_NUM_RELATION 1346
#define KG_DIM          256
#define KG_B            16384
#define KG_NCTX         10
#define KG_PCTX         10
#define KG_PLEN         3

// ---------------- launch configuration ----------------
#define WAVES_PER_BLOCK 2                 // 64 threads/block, 60KB LDS staging
#define BLOCK_THREADS   (WAVES_PER_BLOCK * 32)
#define STAGE_ROWS      30                // max(2*NCTX=20, PCTX*PLEN=30) rows of 1KB
#define MAIN_BLOCKS     (KG_B / WAVES_PER_BLOCK)

// ---------------- CDNA5 async global->LDS gather path ----------------
// The builtin exists on this toolchain (confirmed by round-1 diagnostic); its
// pointer params are 128-bit int-vector pointers in AS1 (global) / AS3 (LDS).
#if defined(__HIP_DEVICE_COMPILE__) && defined(__gfx1250__) && \
    __has_builtin(__builtin_amdgcn_global_load_async_to_lds_b128)
#define HAVE_ASYNC_LDS 1
#else
#define HAVE_ASYNC_LDS 0
#endif

#if HAVE_ASYNC_LDS
typedef int kg_v4i __attribute__((vector_size(4 * sizeof(int))));
typedef __attribute__((address_space(1))) kg_v4i* kg_gptr;
typedef __attribute__((address_space(3))) kg_v4i* kg_lptr;
#endif

// Copy one 1KB embedding row (256 f32) from global to this wave's LDS slot.
// Each lane moves 16B at byte offsets lane*16 and 512+lane*16.
__device__ __forceinline__ void stage_row_1k(const float* __restrict__ gsrc,
                                             float* __restrict__ ldst,
                                             int lane) {
#if HAVE_ASYNC_LDS
  __attribute__((address_space(1))) char* g =
      (__attribute__((address_space(1))) char*)const_cast<float*>(gsrc);
  __attribute__((address_space(3))) char* l =
      (__attribute__((address_space(3))) char*)ldst;
  __builtin_amdgcn_global_load_async_to_lds_b128(
      (kg_gptr)(g + lane * 16), (kg_lptr)(l + lane * 16), 0, 0);
  __builtin_amdgcn_global_load_async_to_lds_b128(
      (kg_gptr)(g + 512 + lane * 16), (kg_lptr)(l + 512 + lane * 16), 0, 0);
#else
  float4 a = *(const float4*)(gsrc + 4 * lane);
  float4 b = *(const float4*)(gsrc + 128 + 4 * lane);
  *(float4*)(ldst + 4 * lane) = a;
  *(float4*)(ldst + 128 + 4 * lane) = b;
#endif
}

__device__ __forceinline__ void wait_async_lds() {
#if HAVE_ASYNC_LDS
  asm volatile("" ::: "memory");
#if __has_builtin(__builtin_amdgcn_s_wait_asynccnt)
  __builtin_amdgcn_s_wait_asynccnt(0);
#else
  asm volatile("s_wait_asynccnt 0" ::: "memory");
#endif
  asm volatile("" ::: "memory");
#endif
}

// ASYNC LDS writes are unordered vs this wave's DS reads: drain DScnt before
// re-staging over a region we just read from.
__device__ __forceinline__ void drain_lds_reads() {
#if HAVE_ASYNC_LDS
  asm volatile("s_wait_dscnt 0" ::: "memory");
#endif
}

// ---------------- per-lane 8-element vector fragment ----------------
// Lane l owns dims {4l..4l+3} U {128+4l..128+4l+3} of each 256-dim vector
// (matches the 16B/lane staging granularity; all ops are elementwise+reduce,
// so any fixed dim->lane map is valid as long as it is consistent).
struct F8 { float v[8]; };

__device__ __forceinline__ F8 load_row8(const float* p, int lane) {
  F8 r;
  float4 a = *(const float4*)(p + 4 * lane);
  float4 b = *(const float4*)(p + 128 + 4 * lane);
  r.v[0] = a.x; r.v[1] = a.y; r.v[2] = a.z; r.v[3] = a.w;
  r.v[4] = b.x; r.v[5] = b.y; r.v[6] = b.z; r.v[7] = b.w;
  return r;
}

__device__ __forceinline__ void store_row8(float* p, int lane, const F8& r) {
  float4 a, b;
  a.x = r.v[0]; a.y = r.v[1]; a.z = r.v[2]; a.w = r.v[3];
  b.x = r.v[4]; b.y = r.v[5]; b.z = r.v[6]; b.w = r.v[7];
  *(float4*)(p + 4 * lane) = a;
  *(float4*)(p + 128 + 4 * lane) = b;
}

__device__ __forceinline__ float wave_sum(float x) {
#pragma unroll
  for (int m = 16; m >= 1; m >>= 1) x += __shfl_xor(x, m, 32);
  return x;  // broadcast to all 32 lanes
}

// ---------------- main kernel: one wave32 per sample ----------------
__launch_bounds__(BLOCK_THREADS)
__global__ void kg_loss_main(
    const float* __restrict__ E, const float* __restrict__ R,
    const int* __restrict__ h, const int* __restrict__ r,
    const int* __restrict__ t, const int* __restrict__ h_neg,
    const int* __restrict__ r_neg, const int* __restrict__ t_neg,
    const int* __restrict__ nctx, const int* __restrict__ pctx,
    float* __restrict__ block_partial) {
  __shared__ float stage[WAVES_PER_BLOCK][STAGE_ROWS * KG_DIM];  // 60KB
  __shared__ float wave_loss[WAVES_PER_BLOCK];

  const int lane = threadIdx.x & 31;
  const int wv   = threadIdx.x >> 5;
  const int i    = blockIdx.x * WAVES_PER_BLOCK + wv;
  float* my = stage[wv];

  const int ih  = h[i],     ir  = r[i],     it  = t[i];
  const int ihn = h_neg[i], irn = r_neg[i], itn = t_neg[i];

  // Positive/negative sample embeddings (note: reference uses E[r_neg]).
  F8 eh  = load_row8(E + (size_t)ih  * KG_DIM, lane);
  F8 er  = load_row8(R + (size_t)ir  * KG_DIM, lane);
  F8 et  = load_row8(E + (size_t)it  * KG_DIM, lane);
  F8 ehn = load_row8(E + (size_t)ihn * KG_DIM, lane);
  F8 ern = load_row8(E + (size_t)irn * KG_DIM, lane);
  F8 etn = load_row8(E + (size_t)itn * KG_DIM, lane);

  // ================= neighbor context =================
  // neighbor_tmp = nt - nr ;  a_j = || (nr - nt) + er - et ||
  const int* nc = nctx + (size_t)i * (KG_NCTX * 2);
#pragma unroll
  for (int j = 0; j < KG_NCTX; ++j) {
    stage_row_1k(R + (size_t)nc[2 * j + 0] * KG_DIM, my + (2 * j + 0) * KG_DIM, lane);
    stage_row_1k(E + (size_t)nc[2 * j + 1] * KG_DIM, my + (2 * j + 1) * KG_DIM, lane);
  }
  wait_async_lds();

  float a[KG_NCTX];
#pragma unroll
  for (int j = 0; j < KG_NCTX; ++j) {
    F8 nr = load_row8(my + (2 * j + 0) * KG_DIM, lane);
    F8 nt = load_row8(my + (2 * j + 1) * KG_DIM, lane);
    float s = 0.f;
#pragma unroll
    for (int k = 0; k < 8; ++k) {
      float d = (nr.v[k] - nt.v[k]) + er.v[k] - et.v[k];
      s = fmaf(d, d, s);
    }
    a[j] = sqrtf(wave_sum(s));
  }
  // softmax(-a): exp(min(a) - a_j) / sum
  float amin = a[0];
#pragma unroll
  for (int j = 1; j < KG_NCTX; ++j) amin = fminf(amin, a[j]);
  float wj[KG_NCTX], den = 0.f;
#pragma unroll
  for (int j = 0; j < KG_NCTX; ++j) { wj[j] = expf(amin - a[j]); den += wj[j]; }
  const float inv_den = 1.f / den;

  F8 nagg;
#pragma unroll
  for (int k = 0; k < 8; ++k) nagg.v[k] = 0.f;
#pragma unroll
  for (int j = 0; j < KG_NCTX; ++j) {
    F8 nr = load_row8(my + (2 * j + 0) * KG_DIM, lane);
    F8 nt = load_row8(my + (2 * j + 1) * KG_DIM, lane);
    const float al = wj[j] * inv_den;
#pragma unroll
    for (int k = 0; k < 8; ++k) nagg.v[k] = fmaf(al, nt.v[k] - nr.v[k], nagg.v[k]);
  }
  float sp = 0.f, sn = 0.f;
#pragma unroll
  for (int k = 0; k < 8; ++k) {
    float d = nagg.v[k] - eh.v[k];  sp = fmaf(d, d, sp);
    float e = nagg.v[k] - ehn.v[k]; sn = fmaf(e, e, sn);
  }
  const float g_n_pos = -sqrtf(wave_sum(sp));
  const float g_n_neg = -sqrtf(wave_sum(sn));

  // ================= path context =================
  const int* pc = pctx + (size_t)i * (KG_PCTX * KG_PLEN);
  float psgn[KG_PCTX * KG_PLEN];
  int   pidx[KG_PCTX * KG_PLEN];
#pragma unroll
  for (int q = 0; q < KG_PCTX * KG_PLEN; ++q) {
    int v = pc[q];
    pidx[q] = (v < 0) ? -v : v;
    psgn[q] = (v > 0) ? 1.f : ((v < 0) ? -1.f : 0.f);
  }

  drain_lds_reads();  // WAR: async writes below reuse the neighbor staging area
#pragma unroll
  for (int q = 0; q < KG_PCTX * KG_PLEN; ++q)
    stage_row_1k(R + (size_t)pidx[q] * KG_DIM, my + q * KG_DIM, lane);
  wait_async_lds();

  float b[KG_PCTX];
#pragma unroll
  for (int p = 0; p < KG_PCTX; ++p) {
    F8 r0 = load_row8(my + (3 * p + 0) * KG_DIM, lane);
    F8 r1 = load_row8(my + (3 * p + 1) * KG_DIM, lane);
    F8 r2 = load_row8(my + (3 * p + 2) * KG_DIM, lane);
    F8 ep;
    float s = 0.f;
#pragma unroll
    for (int k = 0; k < 8; ++k) {
      float e = psgn[3 * p + 0] * r0.v[k] + psgn[3 * p + 1] * r1.v[k] +
                psgn[3 * p + 2] * r2.v[k];
      ep.v[k] = e;
      float d = eh.v[k] + e - et.v[k];
      s = fmaf(d, d, s);
    }
    b[p] = sqrtf(wave_sum(s));
    store_row8(my + (3 * p + 0) * KG_DIM, lane, ep);  // keep embed_path for beta pass
  }
  float bmin = b[0];
#pragma unroll
  for (int p = 1; p < KG_PCTX; ++p) bmin = fminf(bmin, b[p]);
  float bw[KG_PCTX], bden = 0.f;
#pragma unroll
  for (int p = 0; p < KG_PCTX; ++p) { bw[p] = expf(bmin - b[p]); bden += bw[p]; }
  const float inv_bden = 1.f / bden;

  F8 pagg;
#pragma unroll
  for (int k = 0; k < 8; ++k) pagg.v[k] = 0.f;
#pragma unroll
  for (int p = 0; p < KG_PCTX; ++p) {
    F8 ep = load_row8(my + (3 * p + 0) * KG_DIM, lane);
    const float be = bw[p] * inv_bden;
#pragma unroll
    for (int k = 0; k < 8; ++k) pagg.v[k] = fmaf(be, ep.v[k], pagg.v[k]);
  }
  float qp = 0.f, qn = 0.f;
#pragma unroll
  for (int k = 0; k < 8; ++k) {
    float d = pagg.v[k] - er.v[k];  qp = fmaf(d, d, qp);
    float e = pagg.v[k] - ern.v[k]; qn = fmaf(e, e, qn);
  }
  const float g_p_pos = -sqrtf(wave_sum(qp));
  const float g_p_neg = -sqrtf(wave_sum(qn));

  // ================= plain triple term =================
  float tp = 0.f, tn2 = 0.f;
#pragma unroll
  for (int k = 0; k < 8; ++k) {
    float d = eh.v[k] + er.v[k] - et.v[k];    tp  = fmaf(d, d, tp);
    float e = ehn.v[k] + ern.v[k] - etn.v[k]; tn2 = fmaf(e, e, tn2);
  }
  const float g_t_pos = -sqrtf(wave_sum(tp));
  const float g_t_neg = -sqrtf(wave_sum(tn2));

  const float L = fmaxf(0.f, 1.f - (g_n_pos - g_n_neg)) +
                  fmaxf(0.f, 1.f - (g_p_pos - g_p_neg)) +
                  fmaxf(0.f, 1.f - (g_t_pos - g_t_neg));

  if (lane == 0) wave_loss[wv] = L;
  __syncthreads();
  if (threadIdx.x == 0) {
    float s = 0.f;
#pragma unroll
    for (int w = 0; w < WAVES_PER_BLOCK; ++w) s += wave_loss[w];
    block_partial[blockIdx.x] = s;
  }
}

// ---------------- deterministic final reduction ----------------
__launch_bounds__(256)
__global__ void kg_final_reduce(const float* __restrict__ partial,
                                float* __restrict__ out, int n) {
  __shared__ float s[256];
  float acc = 0.f;
  for (int idx = threadIdx.x; idx < n; idx += 256) acc += partial[idx];
  s[threadIdx.x] = acc;
  __syncthreads();
  for (int off = 128; off > 0; off >>= 1) {
    if ((int)threadIdx.x < off) s[threadIdx.x] += s[threadIdx.x + off];
    __syncthreads();
  }
  if (threadIdx.x == 0) out[0] = s[0];
}

// ---------------- entry point ----------------
extern "C" void kernel_launch(void* const* d_in, const int* in_sizes, int n_in,
                              void* d_out, int out_size, void* d_ws,
                              size_t ws_size, hipStream_t stream) {
  (void)in_sizes; (void)n_in; (void)out_size; (void)ws_size;
  const float* E    = (const float*)d_in[0];
  const float* R    = (const float*)d_in[1];
  const int*   h    = (const int*)d_in[2];
  const int*   r    = (const int*)d_in[3];
  const int*   t    = (const int*)d_in[4];
  const int*   hn   = (const int*)d_in[5];
  const int*   rn   = (const int*)d_in[6];
  const int*   tn   = (const int*)d_in[7];
  const int*   nctx = (const int*)d_in[8];
  const int*   pctx = (const int*)d_in[9];
  float* partial = (float*)d_ws;            // MAIN_BLOCKS floats = 32KB
  float* out     = (float*)d_out;

  kg_loss_main<<<MAIN_BLOCKS, BLOCK_THREADS, 0, stream>>>(
      E, R, h, r, t, hn, rn, tn, nctx, pctx, partial);
  kg_final_reduce<<<1, 256, 0, stream>>>(partial, out, MAIN_BLOCKS);
}